// Model_6313601925644
// MI455X (gfx1250) — compile-verified
//
#include <hip/hip_runtime.h>
#include <hip/hip_bf16.h>

typedef __attribute__((ext_vector_type(2))) float v2f;
typedef __attribute__((ext_vector_type(8))) float v8f;

#define B_DIM 64
#define N_DIM 64
#define P_DIM 128
#define K_DIM 16
#define NN    4096   // N*N

// ---------------------------------------------------------------------------
// Kernel 1: wg = (weights^2) @ g  (16 x 4096 = [16,4096]x[4096,4096])
//           E  = exp(-wg)         (workspace, same [16][4096] layout)
// One wave per 16-column output tile -> 256 waves. M=16 fits one WMMA tile.
// fp32 WMMA: V_WMMA_F32_16X16X4_F32, loop k in steps of 4 (1024 iters).
// Memory-bound on streaming g (64 MB @ 23.3 TB/s ~ 2.8us floor).
// ---------------------------------------------------------------------------
__global__ __launch_bounds__(256)
void wg_gemm_kernel(const float* __restrict__ W, const float* __restrict__ G,
                    float* __restrict__ wg_out, float* __restrict__ E_out) {
  const int wave = (blockIdx.x * blockDim.x + threadIdx.x) >> 5;  // 0..255
  const int lane = threadIdx.x & 31;
  const int half = lane >> 4;   // 0: lanes 0-15, 1: lanes 16-31
  const int l    = lane & 15;
  const int c0   = wave * 16;   // output column tile base

  // A (16x4 f32) layout: lanes 0-15 M=0-15 hold K={kk,kk+1}; lanes 16-31 hold K={kk+2,kk+3}
  const float* wrow = W + l * NN;

  v8f acc = {};
  #pragma unroll 4
  for (int kk = 0; kk < NN; kk += 4) {
    const int ka = kk + 2 * half;
    // square weights on load (L2-resident, 256 KB)
    float w0 = wrow[ka];
    float w1 = wrow[ka + 1];
    v2f a; a.x = w0 * w0; a.y = w1 * w1;
    // B (4x16 f32): VGPR0 = rows {ka} lanes0-15 / {ka+2} lanes16-31 at cols c0+l
    v2f b;
    b.x = G[(size_t)ka       * NN + c0 + l];
    b.y = G[(size_t)(ka + 1) * NN + c0 + l];
    // stream-prefetch g ~8 k-steps (32 rows) ahead -> global_prefetch_b8
    int kp = (kk + 32 < NN) ? (kk + 32 + 2 * half) : ka;
    __builtin_prefetch(&G[(size_t)kp * NN + c0 + l], 0, 0);
    acc = __builtin_amdgcn_wmma_f32_16x16x4_f32(false, a, false, b,
                                                (short)0, acc, false, false);
  }

  // C/D layout: VGPR r, lanes 0-15 -> row r, lanes 16-31 -> row r+8; col = c0+l
  #pragma unroll
  for (int r = 0; r < 8; ++r) {
    const int m   = r + half * 8;       // wg row (0..15)
    const int col = c0 + l;             // wg col (0..4095)
    const float v = acc[r];
    wg_out[m * NN + col] = v;
    E_out [m * NN + col] = __expf(-v);
  }
}

// ---------------------------------------------------------------------------
// Kernel 2: S[b][k][n] = sum_{i : mask[b,i]==k} (x[b,n,i] - 0.1*x_i[b,i])
// One block per b. LDS layout s[n*16+k]: for fixed n the 16 mask buckets land
// in 16 distinct banks (stride-16 words), avoiding full serialization.
// ---------------------------------------------------------------------------
__global__ __launch_bounds__(256)
void scatter_sum_kernel(const float* __restrict__ x, const int* __restrict__ mask,
                        const float* __restrict__ x_i, float* __restrict__ S_out) {
  __shared__ float s[N_DIM * K_DIM];
  const int b = blockIdx.x;

  for (int j = threadIdx.x; j < N_DIM * K_DIM; j += blockDim.x) s[j] = 0.0f;
  __syncthreads();

  for (int idx = threadIdx.x; idx < N_DIM * P_DIM; idx += blockDim.x) {
    const int n = idx >> 7;       // / P_DIM
    const int i = idx & (P_DIM - 1);
    const int k = mask[b * P_DIM + i];
    const float val = x[(size_t)b * N_DIM * P_DIM + n * P_DIM + i]
                      - 0.1f * x_i[b * P_DIM + i];
    atomicAdd(&s[n * K_DIM + k], val);     // ds_add_f32
  }
  __syncthreads();

  // write out flat-contraction layout: S_out[b][c], c = k*64 + n
  for (int j = threadIdx.x; j < N_DIM * K_DIM; j += blockDim.x) {
    const int n = j >> 4;
    const int k = j & 15;
    S_out[(size_t)b * (K_DIM * N_DIM) + k * N_DIM + n] = s[j];
  }
}

// ---------------------------------------------------------------------------
// Kernel 3: Z[b][m] = sum_c S[b][c] * E[k(c)][m][n(c)] + 0.1*y_i[b][m]
// Z = S_flat[64,1024] @ E_flat^T[1024,64] as 16 WMMA 16x16 tiles (1 wave each).
// ---------------------------------------------------------------------------
__global__ __launch_bounds__(32)
void z_gemm_kernel(const float* __restrict__ S, const float* __restrict__ E,
                   const float* __restrict__ y_i, float* __restrict__ Z) {
  const int tile = blockIdx.x;          // 0..15
  const int b0   = (tile >> 2) * 16;    // batch-row tile
  const int m0   = (tile & 3)  * 16;    // output-col tile
  const int lane = threadIdx.x & 31;
  const int half = lane >> 4;
  const int l    = lane & 15;

  v8f acc = {};
  #pragma unroll 4
  for (int c = 0; c < K_DIM * N_DIM; c += 4) {
    const int ca = c + 2 * half;
    v2f a;
    a.x = S[(size_t)(b0 + l) * (K_DIM * N_DIM) + ca];
    a.y = S[(size_t)(b0 + l) * (K_DIM * N_DIM) + ca + 1];
    const int k0 = ca >> 6,        n0 = ca & 63;
    const int k1 = (ca + 1) >> 6,  n1 = (ca + 1) & 63;
    v2f bb;
    bb.x = E[k0 * NN + (m0 + l) * N_DIM + n0];
    bb.y = E[k1 * NN + (m0 + l) * N_DIM + n1];
    acc = __builtin_amdgcn_wmma_f32_16x16x4_f32(false, a, false, bb,
                                                (short)0, acc, false, false);
  }

  #pragma unroll
  for (int r = 0; r < 8; ++r) {
    const int br = b0 + r + half * 8;   // batch index
    const int m  = m0 + l;              // output index
    Z[br * N_DIM + m] = acc[r] + 0.1f * y_i[br * N_DIM + m];
  }
}

// ---------------------------------------------------------------------------
// Inputs (setup_inputs order): x, x_mask, x_i, y_i, weights, g
// d_out: Z [64*64] then wg [16*4096], both fp32.
// Workspace: E (65536 f32) @ ws+0, S (65536 f32) @ ws+256KB.  (512 KB total)
// ---------------------------------------------------------------------------
extern "C" void kernel_launch(void* const* d_in, const int* in_sizes, int n_in,
                              void* d_out, int out_size, void* d_ws, size_t ws_size,
                              hipStream_t stream) {
  const float* x      = (const float*)d_in[0];
  const int*   x_mask = (const int*)  d_in[1];
  const float* x_i    = (const float*)d_in[2];
  const float* y_i    = (const float*)d_in[3];
  const float* W      = (const float*)d_in[4];
  const float* G      = (const float*)d_in[5];

  float* Z      = (float*)d_out;                 // 4096 floats
  float* wg_out = (float*)d_out + B_DIM * N_DIM; // 65536 floats

  float* E_ws = (float*)d_ws;                    // 65536 floats
  float* S_ws = (float*)d_ws + K_DIM * NN;       // 65536 floats

  // 256 waves for the big GEMM (32 blocks x 8 waves)
  wg_gemm_kernel<<<32, 256, 0, stream>>>(W, G, wg_out, E_ws);
  // masked column sums, one block per batch
  scatter_sum_kernel<<<B_DIM, 256, 0, stream>>>(x, x_mask, x_i, S_ws);
  // final 64x64 GEMM + bias epilogue
  z_gemm_kernel<<<16, 32, 0, stream>>>(S_ws, E_ws, y_i, Z);
}